// SNN_CNN_Hybrid_35442070126945
// MI455X (gfx1250) — compile-verified
//
#include <hip/hip_runtime.h>
#include <hip/hip_bf16.h>
#include <math.h>

// ---------------------------------------------------------------------------
// SNN-CNN hybrid for MI455X (gfx1250, wave32, WMMA).
//
// Pipeline (5 kernels, all on `stream`):
//  K1: full 16-step layer-1 LIF (conv 5x5 s2) -> packed spike bitmasks (u16)
//  K2: full 16-step layer-2 LIF (conv 3x3 s2) via V_WMMA_F32_16X16X4_F32,
//      mem2 lives in WMMA accumulators across all timesteps
//  K3: decoder conv 3x3 s1 + bias + relu via WMMA f32
//  K4: convT 4x4 s2 + bias + relu via per-phase WMMA f32 GEMM
//  K5: convT 4x4 s2 (2 out ch) + bias + softplus, scalar (M=2 is WMMA-hostile)
// ---------------------------------------------------------------------------

typedef __attribute__((ext_vector_type(2))) float v2f;
typedef __attribute__((ext_vector_type(8))) float v8f;

__device__ __forceinline__ v8f wmma_f32(v2f a, v2f b, v8f c) {
  // D(16x16) = A(16x4,f32) * B(4x16,f32) + C ; args: (neg_a, A, neg_b, B, c_mod, C, reuse_a, reuse_b)
  return __builtin_amdgcn_wmma_f32_16x16x4_f32(false, a, false, b, (short)0, c,
                                               false, false);
}

#define T_STEPS 16
#define BATCH   32
#define H1R 50
#define H1C 184
#define H2R 25
#define H2C 92
#define INR 100
#define INC 368

// ---------------------------------------------------------------------------
// K1: layer-1 LIF over ALL timesteps. One thread per (b,h,w); 16 channel
// membranes in registers; spikes packed into u16 bitmask per pixel per step.
// Exploits ~10% input sparsity with a branch around the 16-channel FMA.
// ---------------------------------------------------------------------------
__global__ void __launch_bounds__(256) k_snn_l1(
    const float* __restrict__ x, const float* __restrict__ w1,
    unsigned short* __restrict__ s1p) {
  __shared__ float wl[25 * 16];  // [tap][channel]
  for (int i = threadIdx.x; i < 400; i += 256) {
    int c = i / 25, ij = i % 25;
    wl[ij * 16 + c] = w1[i];
  }
  __syncthreads();

  int tid = blockIdx.x * 256 + threadIdx.x;        // 1150*256 == 32*50*184 exactly
  int w = tid % H1C;
  int h = (tid / H1C) % H1R;
  int b = tid / (H1C * H1R);

  float m[16];
#pragma unroll
  for (int c = 0; c < 16; ++c) m[c] = 0.0f;

  for (int t = 0; t < T_STEPS; ++t) {
    float acc[16];
#pragma unroll
    for (int c = 0; c < 16; ++c) acc[c] = m[c] * 0.5f;

    const float* xp = x + ((size_t)(t * BATCH + b)) * (INR * INC);
#pragma unroll
    for (int i = 0; i < 5; ++i) {
      int ih = 2 * h + i - 2;
      if (ih < 0 || ih >= INR) continue;
#pragma unroll
      for (int j = 0; j < 5; ++j) {
        int iw = 2 * w + j - 2;
        if (iw < 0 || iw >= INC) continue;
        float xv = xp[ih * INC + iw];
        if (xv != 0.0f) {
          const float* wp = &wl[(i * 5 + j) * 16];
#pragma unroll
          for (int c = 0; c < 16; ++c) acc[c] += xv * wp[c];
        }
      }
    }
    unsigned int bits = 0;
#pragma unroll
    for (int c = 0; c < 16; ++c) {
      bool s = acc[c] >= 1.0f;
      bits |= ((unsigned)s) << c;
      m[c] = s ? 0.0f : acc[c];
    }
    s1p[((size_t)(t * BATCH + b) * H1R + h) * H1C + w] = (unsigned short)bits;
  }
}

// ---------------------------------------------------------------------------
// K2: layer-2 LIF over ALL timesteps via WMMA f32 implicit GEMM.
// GEMM: D[ch][pix] = sum_k W2[ch][k] * S1[k][pix], K = 16ic*3*3 = 144.
// Each wave: 16 pixels (one row of mem2) x 32 channels (2 M-tiles).
// mem2 lives in the two v8f accumulators across all 16 steps.
// ---------------------------------------------------------------------------
__global__ void __launch_bounds__(256) k_snn_l2(
    const unsigned short* __restrict__ s1p, const float* __restrict__ w2,
    float* __restrict__ mem2) {
  __shared__ float w2l[32 * 144];
  __shared__ unsigned short st[8][112];  // per-wave spike stage [3][36]
  for (int i = threadIdx.x; i < 32 * 144; i += 256) w2l[i] = w2[i];
  __syncthreads();

  const int wv = threadIdx.x >> 5;
  const int l  = threadIdx.x & 31;
  const int task = blockIdx.x * 8 + wv;  // 600 blocks * 8 waves == 32*25*6
  const int owt = task % 6;
  const int oh  = (task / 6) % H2R;
  const int b   = task / 150;
  const int ow0 = owt * 16;
  const int n   = l & 15;
  const int kb  = (l >> 4) << 1;  // K pair base: 0 (lanes 0-15) / 2 (lanes 16-31)

  v8f acc0 = {};
  v8f acc1 = {};

  for (int t = 0; t < T_STEPS; ++t) {
    __syncthreads();  // all 8 waves run identical trip counts -> safe
    // stage 3 rows x 33 cols of packed spikes (zero padded)
    for (int e = l; e < 99; e += 32) {
      int r = e / 33, c = e % 33;
      int ih = 2 * oh - 1 + r;
      int iw = 2 * ow0 - 1 + c;
      unsigned short v = 0;
      if (ih >= 0 && ih < H1R && iw >= 0 && iw < H1C)
        v = s1p[((size_t)(t * BATCH + b) * H1R + ih) * H1C + iw];
      st[wv][r * 36 + c] = v;
    }
    __syncthreads();

    acc0 = acc0 * 0.5f;  // membrane decay applied to C before accumulation
    acc1 = acc1 * 0.5f;

    for (int kc = 0; kc < 36; ++kc) {
      int k0 = kc * 4 + kb;
      v2f bf;
      {
        int c1 = k0 / 9, r = k0 % 9;
        unsigned int u = st[wv][(r / 3) * 36 + 2 * n + (r % 3)];
        bf.x = (float)((u >> c1) & 1u);
        int k1 = k0 + 1;
        c1 = k1 / 9; r = k1 % 9;
        u = st[wv][(r / 3) * 36 + 2 * n + (r % 3)];
        bf.y = (float)((u >> c1) & 1u);
      }
      v2f a0, a1;
      a0.x = w2l[n * 144 + k0];
      a0.y = w2l[n * 144 + k0 + 1];
      a1.x = w2l[(n + 16) * 144 + k0];
      a1.y = w2l[(n + 16) * 144 + k0 + 1];
      acc0 = wmma_f32(a0, bf, acc0);
      acc1 = wmma_f32(a1, bf, acc1);
    }
    // LIF threshold + reset, elementwise on accumulator fragments
#pragma unroll
    for (int g = 0; g < 8; ++g) {
      float x0 = acc0[g]; acc0[g] = (x0 >= 1.0f) ? 0.0f : x0;
      float x1 = acc1[g]; acc1[g] = (x1 >= 1.0f) ? 0.0f : x1;
    }
  }

  const int mhalf = (l >> 4) << 3;  // D row offset for lanes 16-31
  const int ow = ow0 + n;
  if (ow < H2C) {
#pragma unroll
    for (int g = 0; g < 8; ++g) {
      int ch0 = g + mhalf;
      int ch1 = 16 + g + mhalf;
      mem2[((size_t)(b * 32 + ch0) * H2R + oh) * H2C + ow] = acc0[g];
      mem2[((size_t)(b * 32 + ch1) * H2R + oh) * H2C + ow] = acc1[g];
    }
  }
}

// ---------------------------------------------------------------------------
// K3: decoder conv 3x3 s1 p1 (32->64) + bias + relu, WMMA f32.
// K = 32*9 = 288; M = 64 channels (4 tiles); N = 16 pixels.
// ---------------------------------------------------------------------------
__global__ void __launch_bounds__(256) k_dec_conv(
    const float* __restrict__ mem2, const float* __restrict__ wd,
    const float* __restrict__ bd, float* __restrict__ h1) {
  const int wvg = blockIdx.x * 8 + (threadIdx.x >> 5);  // 600 blocks
  const int l = threadIdx.x & 31;
  const int owt = wvg % 6;
  const int oh  = (wvg / 6) % H2R;
  const int b   = wvg / 150;
  const int ow0 = owt * 16;
  const int n   = l & 15;
  const int kb  = (l >> 4) << 1;

  v8f acc[4] = {{}, {}, {}, {}};
  for (int kc = 0; kc < 72; ++kc) {
    int k0 = kc * 4 + kb;
    v2f bf;
#pragma unroll
    for (int e = 0; e < 2; ++e) {
      int k = k0 + e;
      int c = k / 9, r = k % 9;
      int ih = oh + r / 3 - 1;
      int iw = ow0 + n + (r % 3) - 1;
      float v = 0.0f;
      if (ih >= 0 && ih < H2R && iw >= 0 && iw < H2C)
        v = mem2[((size_t)(b * 32 + c) * H2R + ih) * H2C + iw];
      if (e == 0) bf.x = v; else bf.y = v;
    }
#pragma unroll
    for (int mt = 0; mt < 4; ++mt) {
      v2f af;
      af.x = wd[(mt * 16 + n) * 288 + k0];
      af.y = wd[(mt * 16 + n) * 288 + k0 + 1];
      acc[mt] = wmma_f32(af, bf, acc[mt]);
    }
  }
  const int mhalf = (l >> 4) << 3;
  const int ow = ow0 + n;
  if (ow < H2C) {
#pragma unroll
    for (int mt = 0; mt < 4; ++mt)
#pragma unroll
      for (int g = 0; g < 8; ++g) {
        int ch = mt * 16 + g + mhalf;
        h1[((size_t)(b * 64 + ch) * H2R + oh) * H2C + ow] =
            fmaxf(acc[mt][g] + bd[ch], 0.0f);
      }
  }
}

// ---------------------------------------------------------------------------
// K4: convT 4x4 s2 p1 (64->32) + bias + relu, per-output-phase WMMA f32 GEMM.
// y[oc][oh][ow] = sum_{ic,kh,kw} h1[ic][ih][iw]*wt1[ic][oc][kh][kw],
// oh = 2*ih + kh - 1. Pixels tiled per (oh, ow-parity) so taps are uniform:
// K = 64ic * 2j * 2i = 256.
// ---------------------------------------------------------------------------
__global__ void __launch_bounds__(256) k_convT1(
    const float* __restrict__ h1, const float* __restrict__ wt1,
    const float* __restrict__ bt1, float* __restrict__ h2) {
  const int wvg = blockIdx.x * 8 + (threadIdx.x >> 5);  // 2400 blocks
  const int l = threadIdx.x & 31;
  const int owt = wvg % 6;
  const int pw  = (wvg / 6) % 2;
  const int oh  = (wvg / 12) % H1R;
  const int b   = wvg / 600;
  const int n   = l & 15;
  const int kb  = (l >> 4) << 1;
  const int ph = oh & 1, qh = oh >> 1;
  const int q0 = owt * 16;

  v8f acc0 = {}, acc1 = {};
  for (int kc = 0; kc < 64; ++kc) {
    int k0 = kc * 4 + kb;
    v2f bf;
#pragma unroll
    for (int e = 0; e < 2; ++e) {
      int k = k0 + e;
      int ic = k >> 2, j = (k >> 1) & 1, i = k & 1;
      int ih = ph + qh - j;
      int iw = pw + q0 + n - i;
      float v = 0.0f;
      if (ih >= 0 && ih < H2R && iw >= 0 && iw < H2C)
        v = h1[((size_t)(b * 64 + ic) * H2R + ih) * H2C + iw];
      if (e == 0) bf.x = v; else bf.y = v;
    }
    // A fragments: weights, m = out channel within tile
    int ic0 = k0 >> 2, j0 = (k0 >> 1) & 1, i0 = k0 & 1;
    int ic1 = (k0 + 1) >> 2, j1 = ((k0 + 1) >> 1) & 1, i1 = (k0 + 1) & 1;
    int kh0 = (1 - ph) + 2 * j0, kw0 = (1 - pw) + 2 * i0;
    int kh1 = (1 - ph) + 2 * j1, kw1 = (1 - pw) + 2 * i1;
    v2f a0, a1;
    a0.x = wt1[ic0 * 512 + n * 16 + kh0 * 4 + kw0];
    a0.y = wt1[ic1 * 512 + n * 16 + kh1 * 4 + kw1];
    a1.x = wt1[ic0 * 512 + (n + 16) * 16 + kh0 * 4 + kw0];
    a1.y = wt1[ic1 * 512 + (n + 16) * 16 + kh1 * 4 + kw1];
    acc0 = wmma_f32(a0, bf, acc0);
    acc1 = wmma_f32(a1, bf, acc1);
  }
  const int mhalf = (l >> 4) << 3;
  const int q = q0 + n;
  if (q < 92) {
    const int ow = pw + 2 * q;
#pragma unroll
    for (int g = 0; g < 8; ++g) {
      int oc0 = g + mhalf;
      int oc1 = 16 + g + mhalf;
      h2[((size_t)(b * 32 + oc0) * H1R + oh) * H1C + ow] =
          fmaxf(acc0[g] + bt1[oc0], 0.0f);
      h2[((size_t)(b * 32 + oc1) * H1R + oh) * H1C + ow] =
          fmaxf(acc1[g] + bt1[oc1], 0.0f);
    }
  }
}

// ---------------------------------------------------------------------------
// K5: convT 4x4 s2 p1 (32->2) + bias + softplus, scalar (M=2).
// Output split per reference tuple: [B,1,100,368] ch0 then [B,1,100,368] ch1.
// ---------------------------------------------------------------------------
__device__ __forceinline__ float softplus_f(float x) {
  return fmaxf(x, 0.0f) + log1pf(expf(-fabsf(x)));
}

__global__ void __launch_bounds__(256) k_convT2(
    const float* __restrict__ h2, const float* __restrict__ wt2,
    const float* __restrict__ bt2, float* __restrict__ out) {
  int tid = blockIdx.x * 256 + threadIdx.x;  // 4600*256 == 32*100*368
  int ow = tid % INC;
  int oh = (tid / INC) % INR;
  int b  = tid / (INC * INR);
  int p = ow & 1, q = ow >> 1;
  int ph = oh & 1, qh = oh >> 1;
  float acc0 = bt2[0], acc1 = bt2[1];
  for (int ic = 0; ic < 32; ++ic) {
    const float* hp = h2 + (size_t)(b * 32 + ic) * (H1R * H1C);
#pragma unroll
    for (int j = 0; j < 2; ++j) {
      int ih = ph + qh - j;
      if (ih < 0 || ih >= H1R) continue;
      int kh = (1 - ph) + 2 * j;
#pragma unroll
      for (int i = 0; i < 2; ++i) {
        int iw = p + q - i;
        if (iw < 0 || iw >= H1C) continue;
        int kw = (1 - p) + 2 * i;
        float xv = hp[ih * H1C + iw];
        acc0 += xv * wt2[ic * 32 + kh * 4 + kw];
        acc1 += xv * wt2[ic * 32 + 16 + kh * 4 + kw];
      }
    }
  }
  size_t plane = (size_t)INR * INC;
  out[(size_t)b * plane + oh * INC + ow] = softplus_f(acc0);
  out[(size_t)BATCH * plane + (size_t)b * plane + oh * INC + ow] = softplus_f(acc1);
}

// ---------------------------------------------------------------------------
extern "C" void kernel_launch(void* const* d_in, const int* in_sizes, int n_in,
                              void* d_out, int out_size, void* d_ws,
                              size_t ws_size, hipStream_t stream) {
  const float* x   = (const float*)d_in[0];
  const float* w1  = (const float*)d_in[1];
  const float* w2  = (const float*)d_in[2];
  const float* wd  = (const float*)d_in[3];
  const float* bd  = (const float*)d_in[4];
  const float* wt1 = (const float*)d_in[5];
  const float* bt1 = (const float*)d_in[6];
  const float* wt2 = (const float*)d_in[7];
  const float* bt2 = (const float*)d_in[8];
  float* out = (float*)d_out;

  char* ws = (char*)d_ws;
  unsigned short* s1p = (unsigned short*)(ws);                 //  9,420,800 B
  float* mem2 = (float*)(ws + 9420800);                        //  9,420,800 B
  float* h1   = (float*)(ws + 18841600);                       // 18,841,600 B
  float* h2   = (float*)(ws + 37683200);                       // 37,683,200 B

  k_snn_l1 <<<1150, 256, 0, stream>>>(x, w1, s1p);
  k_snn_l2 <<<600,  256, 0, stream>>>(s1p, w2, mem2);
  k_dec_conv<<<600, 256, 0, stream>>>(mem2, wd, bd, h1);
  k_convT1 <<<2400, 256, 0, stream>>>(h1, wt1, bt1, h2);
  k_convT2 <<<4600, 256, 0, stream>>>(h2, wt2, bt2, out);
}